// S5Layer_36567351558290
// MI455X (gfx1250) — compile-verified
//
#include <hip/hip_runtime.h>
#include <math.h>

// ---------------------------------------------------------------------------
// S5 layer for MI455X (gfx1250, wave32, WMMA).
//   T=8192, H=512, P=512
//   bx   = x @ b_bar^T            (complex)   -> fp32 WMMA GEMM
//   s_t  = mask ? bx_t : lam*s+bx (scan)      -> 3-pass chunked scan
//   ys   = Re(s) @ Re(c)^T - Im(s) @ Im(c)^T + d*x -> fp32 WMMA GEMM
// ---------------------------------------------------------------------------

#define T_DIM 8192
#define H_DIM 512
#define P_DIM 512
#define CHUNK_LEN 64
#define N_CHUNKS (T_DIM / CHUNK_LEN)          // 128

#define A_PITCH 34            // even pitch -> ds_load_b64-able A fragments
#define B_PITCH 130           // per-k-pair row pitch (floats), conflict-free stores

typedef float v2f __attribute__((ext_vector_type(2)));
typedef float v8f __attribute__((ext_vector_type(8)));

// D = A(16x4 f32) * B(4x16 f32) + C(16x16 f32)
__device__ __forceinline__ v8f wmma4(v2f a, v2f b, v8f c) {
    return __builtin_amdgcn_wmma_f32_16x16x4_f32(
        /*neg_a=*/false, a, /*neg_b=*/false, b,
        /*c_mod=*/(short)0, c, /*reuse_a=*/false, /*reuse_b=*/false);
}

__device__ __forceinline__ v2f lds_ld2(const float* p) {
    return *(const v2f*)p;    // 8B-aligned by construction -> ds_load_b64
}

// ---------------------------------------------------------------------------
// Kernel 0: per-p discretization coefficients
// ---------------------------------------------------------------------------
__global__ void s5_coef(const float* __restrict__ lr, const float* __restrict__ li,
                        const float* __restrict__ log_step,
                        float* __restrict__ lbr, float* __restrict__ lbi,
                        float* __restrict__ cfr, float* __restrict__ cfi) {
    int p = blockIdx.x * blockDim.x + threadIdx.x;
    if (p >= P_DIM) return;
    float lre = fminf(lr[p], -1e-4f);
    float lim = li[p];
    float step = expf(log_step[p]);
    float zr = lre * step, zi = lim * step;
    float mag = expf(zr);
    float br = mag * cosf(zi);
    float bi = mag * sinf(zi);
    lbr[p] = br; lbi[p] = bi;
    float nr = br - 1.0f, ni = bi;
    float den = lre * lre + lim * lim;
    cfr[p] = (nr * lre + ni * lim) / den;
    cfi[p] = (ni * lre - nr * lim) / den;
}

// Kernel 0b: b_bar[p][h] = coef[p] * b_tilde[p][h]   (b layout (P,H,2))
__global__ void s5_bbar(const float* __restrict__ b,
                        const float* __restrict__ cfr, const float* __restrict__ cfi,
                        float* __restrict__ bbr, float* __restrict__ bbi) {
    int idx = blockIdx.x * blockDim.x + threadIdx.x;   // p*H + h
    int h = idx & (H_DIM - 1);
    int p = idx >> 9;
    float btr = b[(size_t)p * (2 * H_DIM) + 2 * h + 0];
    float bti = b[(size_t)p * (2 * H_DIM) + 2 * h + 1];
    float cr = cfr[p], ci = cfi[p];
    bbr[idx] = cr * btr - ci * bti;
    bbi[idx] = cr * bti + ci * btr;
}

// ---------------------------------------------------------------------------
// GEMM 1: bx = x @ b_bar^T  (re & im share the A tile)
//   M=T, N=P, K=H.  Block = 256 thr (8 waves), tile 128x64, Kt=32.
//   B[k][n] = bbar[n*H + k]
// ---------------------------------------------------------------------------
__global__ __launch_bounds__(256) void s5_gemm_bx(
    const float* __restrict__ x,
    const float* __restrict__ bbr, const float* __restrict__ bbi,
    float* __restrict__ bxr, float* __restrict__ bxi) {
    __shared__ __align__(16) float Xs[128 * A_PITCH];
    __shared__ __align__(16) float Br[16 * B_PITCH];
    __shared__ __align__(16) float Bi[16 * B_PITCH];

    const int tid = threadIdx.x;
    const int lane = tid & 31, wave = tid >> 5;
    const int row0 = blockIdx.y * 128;
    const int col0 = blockIdx.x * 64;
    const int r = lane & 15;
    const int kk2 = (lane >> 4) << 1;       // ISA A-layout: lanes16-31 hold K=2,3
    const int wr = wave * 16;

    v8f accR[4], accI[4];
#pragma unroll
    for (int c = 0; c < 4; ++c) { accR[c] = (v8f)0.0f; accI[c] = (v8f)0.0f; }

    const int lkk = tid & 31;
    const int lrow = tid >> 5;
    const int bdst = (lkk >> 1) * B_PITCH + (lkk & 1);   // + 2*n per column

    for (int k0 = 0; k0 < H_DIM; k0 += 32) {
        const float* gx = x + (size_t)row0 * H_DIM + k0;
#pragma unroll
        for (int i = 0; i < 16; ++i)
            Xs[(lrow + i * 8) * A_PITCH + lkk] = gx[(size_t)(lrow + i * 8) * H_DIM + lkk];
#pragma unroll
        for (int i = 0; i < 8; ++i) {
            int n = lrow + i * 8;
            Br[bdst + 2 * n] = bbr[(size_t)(col0 + n) * H_DIM + k0 + lkk];
            Bi[bdst + 2 * n] = bbi[(size_t)(col0 + n) * H_DIM + k0 + lkk];
        }
        __syncthreads();
#pragma unroll
        for (int k = 0; k < 32; k += 4) {
            v2f a = lds_ld2(&Xs[(wr + r) * A_PITCH + k + kk2]);
            const int kp = ((k + kk2) >> 1) * B_PITCH;
#pragma unroll
            for (int c = 0; c < 4; ++c) {
                int n2 = (c * 16 + r) * 2;
                v2f fr = lds_ld2(&Br[kp + n2]);
                v2f fi = lds_ld2(&Bi[kp + n2]);
                accR[c] = wmma4(a, fr, accR[c]);
                accI[c] = wmma4(a, fi, accI[c]);
            }
        }
        __syncthreads();
    }
    const int mb = (lane < 16) ? 0 : 8;     // ISA C/D layout
    const int nn = lane & 15;
#pragma unroll
    for (int c = 0; c < 4; ++c)
#pragma unroll
        for (int rr = 0; rr < 8; ++rr) {
            size_t off = (size_t)(row0 + wr + mb + rr) * P_DIM + (col0 + c * 16 + nn);
            bxr[off] = accR[c][rr];
            bxi[off] = accI[c][rr];
        }
}

// ---------------------------------------------------------------------------
// Scan pass 1: per (chunk,p) fold (A,B,C) summary over 64 steps.
// ---------------------------------------------------------------------------
__global__ void s5_scan1(const float* __restrict__ bxr, const float* __restrict__ bxi,
                         const float* __restrict__ mask,
                         const float* __restrict__ lbr, const float* __restrict__ lbi,
                         float* __restrict__ sAr, float* __restrict__ sAi,
                         float* __restrict__ sBr, float* __restrict__ sBi,
                         float* __restrict__ sCf) {
    int tid = blockIdx.x * blockDim.x + threadIdx.x;   // chunk*P + p
    int p = tid & (P_DIM - 1);
    int chunk = tid >> 9;
    float ar = lbr[p], ai = lbi[p];
    float Ar = 1.f, Ai = 0.f, Br = 0.f, Bi = 0.f, Cf = 0.f;
    int t0 = chunk * CHUNK_LEN;
    for (int i = 0; i < CHUNK_LEN; ++i) {
        int t = t0 + i;
        float m = mask[t];
        float xr = bxr[(size_t)t * P_DIM + p];
        float xi = bxi[(size_t)t * P_DIM + p];
        if (m != 0.0f) {                 // reset: uniform across wave (m depends on t only)
            Ar = ar; Ai = ai; Br = xr; Bi = xi; Cf = 1.f;
        } else {
            float nAr = ar * Ar - ai * Ai, nAi = ar * Ai + ai * Ar;
            float nBr = ar * Br - ai * Bi + xr, nBi = ar * Bi + ai * Br + xi;
            Ar = nAr; Ai = nAi; Br = nBr; Bi = nBi;
        }
    }
    sAr[tid] = Ar; sAi[tid] = Ai; sBr[tid] = Br; sBi[tid] = Bi; sCf[tid] = Cf;
}

// ---------------------------------------------------------------------------
// Scan pass 2: per p, scan 128 chunk summaries; emit per-chunk carry-in and
// the final carry (first 2*P floats of d_out, layout (P,2)).
// ---------------------------------------------------------------------------
__global__ void s5_scan2(const float* __restrict__ carry0,
                         const float* __restrict__ sAr, const float* __restrict__ sAi,
                         const float* __restrict__ sBr, const float* __restrict__ sBi,
                         const float* __restrict__ sCf,
                         float* __restrict__ cinR, float* __restrict__ cinI,
                         float* __restrict__ outCarry) {
    int p = blockIdx.x * blockDim.x + threadIdx.x;
    if (p >= P_DIM) return;
    float sr = carry0[2 * p + 0];
    float si = carry0[2 * p + 1];
    for (int c = 0; c < N_CHUNKS; ++c) {
        int idx = c * P_DIM + p;
        cinR[idx] = sr; cinI[idx] = si;
        float Ar = sAr[idx], Ai = sAi[idx], Br = sBr[idx], Bi = sBi[idx];
        float omc = 1.0f - sCf[idx];
        float tr = Ar * sr - Ai * si;
        float ti = Ar * si + Ai * sr;
        sr = tr * omc + Br;
        si = ti * omc + Bi;
    }
    outCarry[2 * p + 0] = sr;
    outCarry[2 * p + 1] = si;
}

// ---------------------------------------------------------------------------
// Scan pass 3: replay chunks with carry-in, write states in-place over bx.
// ---------------------------------------------------------------------------
__global__ void s5_scan3(float* __restrict__ bxr, float* __restrict__ bxi,
                         const float* __restrict__ mask,
                         const float* __restrict__ lbr, const float* __restrict__ lbi,
                         const float* __restrict__ cinR, const float* __restrict__ cinI) {
    int tid = blockIdx.x * blockDim.x + threadIdx.x;
    int p = tid & (P_DIM - 1);
    int chunk = tid >> 9;
    float ar = lbr[p], ai = lbi[p];
    float sr = cinR[tid], si = cinI[tid];
    int t0 = chunk * CHUNK_LEN;
    for (int i = 0; i < CHUNK_LEN; ++i) {
        int t = t0 + i;
        float m = mask[t];
        size_t off = (size_t)t * P_DIM + p;
        float xr = bxr[off], xi = bxi[off];
        if (m != 0.0f) {
            sr = xr; si = xi;
        } else {
            float nr = ar * sr - ai * si + xr;
            float ni = ar * si + ai * sr + xi;
            sr = nr; si = ni;
        }
        bxr[off] = sr; bxi[off] = si;
    }
}

// ---------------------------------------------------------------------------
// GEMM 2: ys = Sre @ Cre^T - Sim @ Cim^T + d*x   (M=T, N=H, K=P)
//   c layout (H,P,2): B_re[k][n] = c[n*2P + 2k], B_im[k][n] = c[n*2P + 2k+1]
// ---------------------------------------------------------------------------
__global__ __launch_bounds__(256) void s5_gemm_ys(
    const float* __restrict__ sre, const float* __restrict__ sim,
    const float* __restrict__ cmat, const float* __restrict__ dvec,
    const float* __restrict__ x, float* __restrict__ ys) {
    __shared__ __align__(16) float Sr[128 * A_PITCH];
    __shared__ __align__(16) float Si[128 * A_PITCH];
    __shared__ __align__(16) float Cr[16 * B_PITCH];
    __shared__ __align__(16) float Ci[16 * B_PITCH];

    const int tid = threadIdx.x;
    const int lane = tid & 31, wave = tid >> 5;
    const int row0 = blockIdx.y * 128;
    const int col0 = blockIdx.x * 64;
    const int r = lane & 15;
    const int kk2 = (lane >> 4) << 1;
    const int wr = wave * 16;

    v8f accP[4], accM[4];
#pragma unroll
    for (int c = 0; c < 4; ++c) { accP[c] = (v8f)0.0f; accM[c] = (v8f)0.0f; }

    const int lkk = tid & 31;
    const int lrow = tid >> 5;
    const int bdst = (lkk >> 1) * B_PITCH + (lkk & 1);

    for (int k0 = 0; k0 < P_DIM; k0 += 32) {
        const float* gr = sre + (size_t)row0 * P_DIM + k0;
        const float* gi = sim + (size_t)row0 * P_DIM + k0;
#pragma unroll
        for (int i = 0; i < 16; ++i) {
            Sr[(lrow + i * 8) * A_PITCH + lkk] = gr[(size_t)(lrow + i * 8) * P_DIM + lkk];
            Si[(lrow + i * 8) * A_PITCH + lkk] = gi[(size_t)(lrow + i * 8) * P_DIM + lkk];
        }
#pragma unroll
        for (int i = 0; i < 8; ++i) {
            int n = lrow + i * 8;
            size_t cb = (size_t)(col0 + n) * (2 * P_DIM) + 2 * (k0 + lkk);
            Cr[bdst + 2 * n] = cmat[cb + 0];
            Ci[bdst + 2 * n] = cmat[cb + 1];
        }
        __syncthreads();
#pragma unroll
        for (int k = 0; k < 32; k += 4) {
            v2f areg = lds_ld2(&Sr[(wr + r) * A_PITCH + k + kk2]);
            v2f aim  = lds_ld2(&Si[(wr + r) * A_PITCH + k + kk2]);
            const int kp = ((k + kk2) >> 1) * B_PITCH;
#pragma unroll
            for (int c = 0; c < 4; ++c) {
                int n2 = (c * 16 + r) * 2;
                v2f fr = lds_ld2(&Cr[kp + n2]);
                v2f fi = lds_ld2(&Ci[kp + n2]);
                accP[c] = wmma4(areg, fr, accP[c]);
                accM[c] = wmma4(aim, fi, accM[c]);
            }
        }
        __syncthreads();
    }
    const int mb = (lane < 16) ? 0 : 8;
    const int nn = lane & 15;
#pragma unroll
    for (int c = 0; c < 4; ++c) {
        int col = col0 + c * 16 + nn;
        float dv = dvec[col];
#pragma unroll
        for (int rr = 0; rr < 8; ++rr) {
            size_t row = (size_t)(row0 + wr + mb + rr);
            float v = accP[c][rr] - accM[c][rr] + dv * x[row * H_DIM + col];
            ys[row * H_DIM + col] = v;
        }
    }
}

// ---------------------------------------------------------------------------
// Launch
// ---------------------------------------------------------------------------
extern "C" void kernel_launch(void* const* d_in, const int* in_sizes, int n_in,
                              void* d_out, int out_size, void* d_ws, size_t ws_size,
                              hipStream_t stream) {
    const float* x        = (const float*)d_in[0];   // (T,H)
    const float* mask     = (const float*)d_in[1];   // (T,)
    const float* carry0   = (const float*)d_in[2];   // (1,P,2)
    const float* lam_re   = (const float*)d_in[3];   // (P,)
    const float* lam_im   = (const float*)d_in[4];   // (P,)
    const float* b        = (const float*)d_in[5];   // (P,H,2)
    const float* cmat     = (const float*)d_in[6];   // (H,P,2)
    const float* dvec     = (const float*)d_in[7];   // (H,)
    const float* log_step = (const float*)d_in[8];   // (P,1)

    float* outCarry = (float*)d_out;                 // 2*P floats
    float* ys       = (float*)d_out + 2 * P_DIM;     // T*H floats

    // workspace layout (floats)
    const size_t PH = (size_t)P_DIM * H_DIM;
    const size_t TP = (size_t)T_DIM * P_DIM;
    const size_t CP = (size_t)N_CHUNKS * P_DIM;
    float* ws   = (float*)d_ws;
    float* bbr  = ws;                  // P*H
    float* bbi  = bbr + PH;            // P*H
    float* lbr  = bbi + PH;            // P
    float* lbi  = lbr + P_DIM;
    float* cfr  = lbi + P_DIM;
    float* cfi  = cfr + P_DIM;
    float* bxr  = cfi + P_DIM;         // T*P  (becomes states_re)
    float* bxi  = bxr + TP;            // T*P  (becomes states_im)
    float* sAr  = bxi + TP;            // N_CHUNKS*P each
    float* sAi  = sAr + CP;
    float* sBr  = sAi + CP;
    float* sBi  = sBr + CP;
    float* sCf  = sBi + CP;
    float* cinR = sCf + CP;
    float* cinI = cinR + CP;

    s5_coef<<<(P_DIM + 255) / 256, 256, 0, stream>>>(lam_re, lam_im, log_step,
                                                     lbr, lbi, cfr, cfi);
    s5_bbar<<<(unsigned)(PH / 256), 256, 0, stream>>>(b, cfr, cfi, bbr, bbi);

    dim3 g1(P_DIM / 64, T_DIM / 128);
    s5_gemm_bx<<<g1, 256, 0, stream>>>(x, bbr, bbi, bxr, bxi);

    s5_scan1<<<(unsigned)(CP / 256), 256, 0, stream>>>(bxr, bxi, mask, lbr, lbi,
                                                       sAr, sAi, sBr, sBi, sCf);
    s5_scan2<<<(P_DIM + 255) / 256, 256, 0, stream>>>(carry0, sAr, sAi, sBr, sBi, sCf,
                                                      cinR, cinI, outCarry);
    s5_scan3<<<(unsigned)(CP / 256), 256, 0, stream>>>(bxr, bxi, mask, lbr, lbi,
                                                       cinR, cinI);

    dim3 g2(H_DIM / 64, T_DIM / 128);
    s5_gemm_ys<<<g2, 256, 0, stream>>>(bxr, bxi, cmat, dvec, x, ys);
}